// FSSA_12214886990462
// MI455X (gfx1250) — compile-verified
//
#include <hip/hip_runtime.h>
#include <math.h>

// ---------------------------------------------------------------------------
// FSSA (frequency + spatial self-attention) for MI455X / gfx1250.
// B=8, DIM=128, HEADS=8, hd=16, H=W=128, HW=16384.
// All f32. WMMA f32 16x16x4 for attention Grams, attn@V and 1x1-conv GEMMs.
// Async global->LDS staging (gfx1250 ASYNCcnt path) in the GEMM when available.
// ---------------------------------------------------------------------------

typedef __attribute__((ext_vector_type(2))) float v2f;
typedef __attribute__((ext_vector_type(4))) float v4f;
typedef __attribute__((ext_vector_type(8))) float v8f;
typedef __attribute__((ext_vector_type(4))) int   v4i;

#define NB   8
#define NC   128
#define NH   8          // heads
#define HD   16         // head dim
#define HW   16384      // 128*128
#define NI   16777216L  // NB*NC*HW

__device__ __forceinline__ v8f wmma4(v2f a, v2f b, v8f c) {
  // D = A(16x4) * B(4x16) + C(16x16), f32.
  return __builtin_amdgcn_wmma_f32_16x16x4_f32(false, a, false, b, (short)0, c,
                                               false, false);
}

// --------------------- async global->LDS staging helpers -------------------
#if defined(__AMDGCN__) && \
    __has_builtin(__builtin_amdgcn_global_load_async_to_lds_b32) && \
    __has_builtin(__builtin_amdgcn_s_wait_asynccnt)
#define USE_ASYNC_LDS 1
#else
#define USE_ASYNC_LDS 0
#endif

#if USE_ASYNC_LDS && __has_builtin(__builtin_amdgcn_global_load_async_to_lds_b128)
#define USE_ASYNC_B128 1
#else
#define USE_ASYNC_B128 0
#endif

typedef __attribute__((address_space(1))) int as1_int;
typedef __attribute__((address_space(3))) int as3_int;
typedef __attribute__((address_space(1))) v4i as1_v4i;
typedef __attribute__((address_space(3))) v4i as3_v4i;

// copy 4 consecutive floats (16B-aligned on both sides) global -> LDS
__device__ __forceinline__ void stage_b128(float* lds_dst, const float* gsrc) {
#if USE_ASYNC_B128
  __builtin_amdgcn_global_load_async_to_lds_b128((as1_v4i*)gsrc, (as3_v4i*)lds_dst,
                                                 0, 0);
#elif USE_ASYNC_LDS
  __builtin_amdgcn_global_load_async_to_lds_b32((as1_int*)(gsrc + 0),
                                                (as3_int*)(lds_dst + 0), 0, 0);
  __builtin_amdgcn_global_load_async_to_lds_b32((as1_int*)(gsrc + 1),
                                                (as3_int*)(lds_dst + 1), 0, 0);
  __builtin_amdgcn_global_load_async_to_lds_b32((as1_int*)(gsrc + 2),
                                                (as3_int*)(lds_dst + 2), 0, 0);
  __builtin_amdgcn_global_load_async_to_lds_b32((as1_int*)(gsrc + 3),
                                                (as3_int*)(lds_dst + 3), 0, 0);
#else
  *(v4f*)lds_dst = *(const v4f*)gsrc;
#endif
}

__device__ __forceinline__ void stage_wait() {
#if USE_ASYNC_LDS
  __builtin_amdgcn_s_wait_asynccnt(0);
#endif
  __syncthreads();
}

// ---------------------------------------------------------------------------
// Generic 128-point FFT along "lines" of a [img][128][128] layout.
//   line l -> img = l>>7, w = l&127 ; base = img*16384 + w*line_pitch
// rows pass: line_pitch=128, elem_stride=1 ; cols pass: line_pitch=1, stride=128
// sign = -1 forward, +1 inverse; scale applied on store.
// HAS_IM: input has imaginary plane. GATED: multiply input by gate[a].
// ABS: store |z| to out_re only.
// 256 threads = 4 lines/block, 64 butterfly lanes per line.
// ---------------------------------------------------------------------------
template <bool HAS_IM, bool GATED, bool ABS>
__global__ void fft128_kernel(const float* __restrict__ in_re,
                              const float* __restrict__ in_im,
                              const float* __restrict__ gate,
                              float* __restrict__ out_re,
                              float* __restrict__ out_im,
                              int n_lines, int line_pitch, int elem_stride,
                              float sign, float scale) {
  __shared__ float sre[4][128];
  __shared__ float sim[4][128];
  const int tid = threadIdx.x;
  const int r = tid >> 6;
  const int t = tid & 63;
  const int line = blockIdx.x * 4 + r;
  if (line >= n_lines) return;   // grids are exact multiples of 4
  const int img = line >> 7;
  const int within = line & 127;
  const long base = (long)img * 16384 + (long)within * line_pitch;

  for (int j = t; j < 128; j += 64) {
    int src = (int)(__brev((unsigned)j) >> 25);
    long a = base + (long)src * elem_stride;
    float re = in_re[a];
    float im = HAS_IM ? in_im[a] : 0.0f;
    if (GATED) { float g = gate[a]; re *= g; im *= g; }
    sre[r][j] = re;
    sim[r][j] = im;
  }
  __syncthreads();

  for (int s = 1; s <= 7; ++s) {
    const int m = 1 << s, half = m >> 1;
    const int g = t / half, j = t % half;
    const int i0 = g * m + j, i1 = i0 + half;
    float ang = sign * 6.2831853071795864f * (float)j / (float)m;
    float wi, wr;
    __sincosf(ang, &wi, &wr);
    float ur = sre[r][i0], ui = sim[r][i0];
    float vr = sre[r][i1], vi = sim[r][i1];
    float tr = vr * wr - vi * wi;
    float ti = vr * wi + vi * wr;
    sre[r][i0] = ur + tr; sim[r][i0] = ui + ti;
    sre[r][i1] = ur - tr; sim[r][i1] = ui - ti;
    __syncthreads();
  }

  for (int j = t; j < 128; j += 64) {
    long a = base + (long)j * elem_stride;
    float re = sre[r][j] * scale, im = sim[r][j] * scale;
    if (ABS) out_re[a] = sqrtf(re * re + im * im);
    else     { out_re[a] = re; out_im[a] = im; }
  }
}

// ---------------------------------------------------------------------------
// 16384-point IFFT, one line per block, in place. 256 threads.
// 128 KB dynamic LDS (gfx1250: 320 KB/WGP makes a 2^14 FFT fit in one block).
// ---------------------------------------------------------------------------
__global__ void ifft16384_kernel(float* __restrict__ re, float* __restrict__ im) {
  extern __shared__ float lds[];
  float* sre = lds;
  float* sim = lds + 16384;
  const int t = threadIdx.x;
  const long base = (long)blockIdx.x * 16384;

  for (int j = t; j < 16384; j += 256) {
    int src = (int)(__brev((unsigned)j) >> 18);
    sre[j] = re[base + src];
    sim[j] = im[base + src];
  }
  __syncthreads();

  for (int s = 1; s <= 14; ++s) {
    const int m = 1 << s, half = m >> 1;
    for (int bt = t; bt < 8192; bt += 256) {
      const int g = bt / half, j = bt % half;
      const int i0 = g * m + j, i1 = i0 + half;
      float ang = 6.2831853071795864f * (float)j / (float)m;  // inverse sign
      float wi, wr;
      __sincosf(ang, &wi, &wr);
      float ur = sre[i0], ui = sim[i0];
      float vr = sre[i1], vi = sim[i1];
      float tr = vr * wr - vi * wi;
      float ti = vr * wi + vi * wr;
      sre[i0] = ur + tr; sim[i0] = ui + ti;
      sre[i1] = ur - tr; sim[i1] = ui - ti;
    }
    __syncthreads();
  }

  const float sc = 1.0f / 16384.0f;
  for (int j = t; j < 16384; j += 256) {
    re[base + j] = sre[j] * sc;
    im[base + j] = sim[j] * sc;
  }
}

// ---------------------------------------------------------------------------
// 16-point IDFT along head-dim of [bh][16][16384] + abs -> out[b,128,HW]
// ---------------------------------------------------------------------------
__global__ void ifft16_abs_kernel(const float* __restrict__ re,
                                  const float* __restrict__ im,
                                  float* __restrict__ out) {
  long idx = (long)blockIdx.x * blockDim.x + threadIdx.x;  // 64*16384
  if (idx >= (long)64 * 16384) return;
  const int n = (int)(idx & 16383);
  const int bh = (int)(idx >> 14);
  const long base = (long)bh * (16 * 16384) + n;
  float xr[16], xi[16];
  for (int c = 0; c < 16; ++c) {
    xr[c] = re[base + (long)c * 16384];
    xi[c] = im[base + (long)c * 16384];
  }
  const float sc = 1.0f / 16.0f;
  const int b = bh >> 3, head = bh & 7;
  for (int k = 0; k < 16; ++k) {
    float ar = 0.0f, ai = 0.0f;
    for (int nn = 0; nn < 16; ++nn) {
      float ang = 6.2831853071795864f * (float)((k * nn) & 15) / 16.0f;
      float s, c;
      __sincosf(ang, &s, &c);
      ar += xr[nn] * c - xi[nn] * s;
      ai += xr[nn] * s + xi[nn] * c;
    }
    ar *= sc; ai *= sc;
    out[(long)b * NC * HW + (long)(head * 16 + k) * HW + n] = sqrtf(ar * ar + ai * ai);
  }
}

// ---------------------------------------------------------------------------
// L2 norms of 1024 lines of 16384 (optionally complex). Stores 1/max(norm,eps).
// ---------------------------------------------------------------------------
template <bool CPLX>
__global__ void l2norm_kernel(const float* __restrict__ re,
                              const float* __restrict__ im,
                              float* __restrict__ inv_out) {
  __shared__ float red[256];
  const int l = blockIdx.x;
  const long base = (long)l * 16384;
  float s = 0.0f;
  for (int j = threadIdx.x; j < 16384; j += 256) {
    float r = re[base + j];
    s += r * r;
    if (CPLX) { float ii = im[base + j]; s += ii * ii; }
  }
  red[threadIdx.x] = s;
  __syncthreads();
  for (int off = 128; off > 0; off >>= 1) {
    if (threadIdx.x < off) red[threadIdx.x] += red[threadIdx.x + off];
    __syncthreads();
  }
  if (threadIdx.x == 0) {
    float n = fmaxf(sqrtf(red[0]), 1e-12f);
    inv_out[l] = 1.0f / n;
  }
}

// ---------------------------------------------------------------------------
// Gram via WMMA: out[bh][c][d] = temp[h]*inva[c]*invb[d] * sum_k A[c,k]*B[d,k]
// CPLX: full complex Gram. SAME: B==A (self-Gram): exploit ir == ri^T
// (3 WMMA chains + half the loads instead of 4 chains).
// One block (8 waves) per bh; each wave covers 2048 of K; LDS tile reduce.
// ---------------------------------------------------------------------------
template <bool CPLX, bool SAME>
__global__ void gram_wmma(const float* __restrict__ are, const float* __restrict__ aim,
                          const float* __restrict__ bre, const float* __restrict__ bim,
                          const float* __restrict__ inva, const float* __restrict__ invb,
                          const float* __restrict__ temp,
                          float* __restrict__ out_re, float* __restrict__ out_im) {
  __shared__ float red[8][4][256];   // fixed max size: keeps all indices in-bounds
  const int bh = blockIdx.x;
  const int wave = threadIdx.x >> 5;
  const int lane = threadIdx.x & 31;
  const int row = lane & 15;
  const int koff = (lane >> 4) * 2;
  const long base = (long)bh * HD * HW;

  v8f rr0 = {}, rr1 = {}, ii = {}, ri = {}, ir = {};
  const int kstart = wave * 2048, kend = kstart + 2048;
  if (CPLX) {
    for (int k0 = kstart; k0 < kend; k0 += 4) {
      long a = base + (long)row * HW + k0 + koff;
      v2f Ar = {are[a], are[a + 1]};
      v2f Ai = {aim[a], aim[a + 1]};
      if (SAME) {
        rr0 = wmma4(Ar, Ar, rr0);
        ii  = wmma4(Ai, Ai, ii);
        ri  = wmma4(Ar, Ai, ri);
      } else {
        v2f Br = {bre[a], bre[a + 1]};
        v2f Bi = {bim[a], bim[a + 1]};
        rr0 = wmma4(Ar, Br, rr0);
        ii  = wmma4(Ai, Bi, ii);
        ri  = wmma4(Ar, Bi, ri);
        ir  = wmma4(Ai, Br, ir);
      }
    }
  } else {
    // two independent accumulator chains to break the WMMA RAW chain
    for (int k0 = kstart; k0 < kend; k0 += 8) {
      long a = base + (long)row * HW + k0 + koff;
      v2f A0 = {are[a], are[a + 1]};
      v2f B0 = {bre[a], bre[a + 1]};
      rr0 = wmma4(A0, B0, rr0);
      v2f A1 = {are[a + 4], are[a + 5]};
      v2f B1 = {bre[a + 4], bre[a + 5]};
      rr1 = wmma4(A1, B1, rr1);
    }
    rr0 = rr0 + rr1;
  }

  // scatter 16x16 partial tiles to LDS (C/D layout: vgpr r -> M=r(+8), lane&15 -> N)
  const int cb = (lane >> 4) * 8;
  const int d0 = lane & 15;
  for (int r = 0; r < 8; ++r) {
    int e = (cb + r) * 16 + d0;
    red[wave][0][e] = rr0[r];
    if (CPLX) {
      red[wave][1][e] = ii[r];
      red[wave][2][e] = ri[r];
      if (!SAME) red[wave][3][e] = ir[r];
    }
  }
  __syncthreads();
  const int e = threadIdx.x;          // (c,d) = (e>>4, e&15)
  const int c = e >> 4, d = e & 15;
  float srr = 0, sii = 0, sim_ = 0;
  for (int w = 0; w < 8; ++w) {
    srr += red[w][0][e];
    if (CPLX) {
      sii += red[w][1][e];
      if (SAME) sim_ += red[w][2][c * 16 + d] + red[w][2][d * 16 + c];  // ri + ri^T
      else      sim_ += red[w][2][e] + red[w][3][e];
    }
  }
  const float scale = temp[bh & 7] * inva[bh * 16 + c] * invb[bh * 16 + d];
  const long o = (long)bh * 256 + e;
  out_re[o] = (CPLX ? (srr - sii) : srr) * scale;
  if (CPLX) out_im[o] = sim_ * scale;
}

// ---------------------------------------------------------------------------
// softmax over last dim (16); complex mode does re and im independently.
// ---------------------------------------------------------------------------
template <bool CPLX>
__global__ void softmax16_kernel(float* __restrict__ re, float* __restrict__ im,
                                 int n_rows) {
  const int rI = blockIdx.x * blockDim.x + threadIdx.x;
  if (rI >= n_rows) return;
  const long base = (long)rI * 16;
  const int nparts = CPLX ? 2 : 1;
  for (int part = 0; part < nparts; ++part) {
    float* p = part ? im : re;
    float mx = -1e30f;
    for (int j = 0; j < 16; ++j) mx = fmaxf(mx, p[base + j]);
    float e[16], s = 0.0f;
    for (int j = 0; j < 16; ++j) { e[j] = __expf(p[base + j] - mx); s += e[j]; }
    float inv = 1.0f / s;
    for (int j = 0; j < 16; ++j) p[base + j] = e[j] * inv;
  }
}

// ---------------------------------------------------------------------------
// D[bh][c][n] = sum_k attn[bh][c][k] * V[bh][k][n]   (complex if CPLX)
// grid (64, 128), block 256: wave -> one 16-wide N tile; K=16 in 4 WMMA chunks.
// ---------------------------------------------------------------------------
template <bool CPLX>
__global__ void attnv_wmma(const float* __restrict__ are, const float* __restrict__ aim,
                           const float* __restrict__ vre, const float* __restrict__ vim,
                           float* __restrict__ dre, float* __restrict__ dim_) {
  const int bh = blockIdx.x;
  const int wave = threadIdx.x >> 5;
  const int lane = threadIdx.x & 31;
  const int n0 = (blockIdx.y * 8 + wave) * 16;
  const int row = lane & 15;
  const int koff = (lane >> 4) * 2;
  const int col = lane & 15;
  const long abase = (long)bh * 256;
  const long vbase = (long)bh * HD * HW;

  v8f drr = {}, dii = {}, dri = {}, dir = {};
  for (int k0 = 0; k0 < 16; k0 += 4) {
    const int kk = k0 + koff;
    v2f Ar = {are[abase + row * 16 + kk], are[abase + row * 16 + kk + 1]};
    const long vb = vbase + (long)kk * HW + n0 + col;
    v2f Br = {vre[vb], vre[vb + HW]};
    drr = wmma4(Ar, Br, drr);
    if (CPLX) {
      v2f Ai = {aim[abase + row * 16 + kk], aim[abase + row * 16 + kk + 1]};
      v2f Bi = {vim[vb], vim[vb + HW]};
      dii = wmma4(Ai, Bi, dii);
      dri = wmma4(Ar, Bi, dri);
      dir = wmma4(Ai, Br, dir);
    }
  }
  const int cb = (lane >> 4) * 8;
  for (int r = 0; r < 8; ++r) {
    const long o = vbase + (long)(cb + r) * HW + n0 + col;
    if (CPLX) { dre[o] = drr[r] - dii[r]; dim_[o] = dri[r] + dir[r]; }
    else      { dre[o] = drr[r]; }
  }
}

// ---------------------------------------------------------------------------
// 1x1 conv as WMMA GEMM: out[b,co,n] = (ADD?out:0)+bias[co]+sum_ci W[co,ci]*IN[ci,n]
// IN = in0 (128 ch) or concat(in0,in1) (SPLIT: 256 ch). Cout = 128.
// The 16-wide B pixel panel (Cin x 16) is staged once per block through LDS
// (async b128 global->LDS when the toolchain exposes it) and shared by all 8
// waves. grid (B, 1024), block 256: wave w -> output rows 16w..16w+15.
// ---------------------------------------------------------------------------
template <bool SPLIT, bool HAS_BIAS, bool ADD>
__global__ void gemm1x1_wmma(const float* __restrict__ W, const float* __restrict__ bias,
                             const float* __restrict__ in0,
                             const float* __restrict__ in1,
                             float* __restrict__ out) {
  constexpr int Cin = SPLIT ? 256 : 128;
  __shared__ float sB[Cin * 16];
  const int b = blockIdx.x;
  const int n0 = blockIdx.y * 16;
  const int tid = threadIdx.x;

  // stage B panel IN[ci][n0..n0+15] as 16B vectors (both sides 16B-aligned)
  for (int idx = tid; idx < Cin * 4; idx += 256) {
    const int ci = idx >> 2;
    const int q = (idx & 3) * 4;
    const float* src = (SPLIT && ci >= 128)
        ? (in1 + (long)b * 128 * HW + (long)(ci - 128) * HW + n0 + q)
        : (in0 + (long)b * 128 * HW + (long)ci * HW + n0 + q);
    stage_b128(&sB[ci * 16 + q], src);
  }
  stage_wait();

  const int wave = tid >> 5;
  const int lane = tid & 31;
  const int row = lane & 15;
  const int koff = (lane >> 4) * 2;
  const int col = lane & 15;
  const int m0 = wave * 16;
  const float* wrow = W + (long)(m0 + row) * Cin;

  v8f acc0 = {}, acc1 = {};
  for (int k0 = 0; k0 < Cin; k0 += 8) {
    const int kk = k0 + koff;
    v2f A0 = {wrow[kk], wrow[kk + 1]};
    v2f B0 = {sB[kk * 16 + col], sB[(kk + 1) * 16 + col]};
    acc0 = wmma4(A0, B0, acc0);
    const int kk1 = kk + 4;
    v2f A1 = {wrow[kk1], wrow[kk1 + 1]};
    v2f B1 = {sB[kk1 * 16 + col], sB[(kk1 + 1) * 16 + col]};
    acc1 = wmma4(A1, B1, acc1);
  }
  v8f acc = acc0 + acc1;

  const int cb = (lane >> 4) * 8;
  for (int r = 0; r < 8; ++r) {
    const int co = m0 + cb + r;
    const long o = (long)b * NC * HW + (long)co * HW + n0 + col;
    float v = acc[r];
    if (HAS_BIAS) v += bias[co];
    if (ADD) out[o] = out[o] + v;
    else     out[o] = v;
  }
}

// ---------------------------------------------------------------------------
// grouped convs: out ch o (0..63) = 3x3(pad1) on in ch {2o,2o+1};
//                out ch 64+o      = 5x5(pad2) on in ch {2o,2o+1}
// ---------------------------------------------------------------------------
__global__ void grouped_conv35(const float* __restrict__ in,
                               const float* __restrict__ w3,
                               const float* __restrict__ w5,
                               float* __restrict__ out) {
  long idx = (long)blockIdx.x * blockDim.x + threadIdx.x;  // 8*64*16384
  if (idx >= (long)NB * 64 * HW) return;
  const int n = (int)(idx & 16383);
  const int o = (int)((idx >> 14) & 63);
  const int b = (int)(idx >> 20);
  const int y = n >> 7, x = n & 127;
  const float* p0 = in + (long)b * NC * HW + (long)(2 * o) * HW;
  const float* p1 = p0 + HW;
  const float* k3 = w3 + o * 18;
  const float* k5 = w5 + o * 50;
  float s3 = 0.0f, s5 = 0.0f;
  for (int dy = -2; dy <= 2; ++dy) {
    const int yy = y + dy;
    if (yy < 0 || yy > 127) continue;
    for (int dx = -2; dx <= 2; ++dx) {
      const int xx = x + dx;
      if (xx < 0 || xx > 127) continue;
      const float a0 = p0[yy * 128 + xx];
      const float a1 = p1[yy * 128 + xx];
      s5 += a0 * k5[(dy + 2) * 5 + dx + 2] + a1 * k5[25 + (dy + 2) * 5 + dx + 2];
      if (dy >= -1 && dy <= 1 && dx >= -1 && dx <= 1)
        s3 += a0 * k3[(dy + 1) * 3 + dx + 1] + a1 * k3[9 + (dy + 1) * 3 + dx + 1];
    }
  }
  out[(long)b * NC * HW + (long)o * HW + n] = s3;
  out[(long)b * NC * HW + (long)(64 + o) * HW + n] = s5;
}

// ---------------------------------------------------------------------------
// SE gate: g = sigmoid(W2 * relu(BN(W1 * xr + b1)) + b2) per pixel (128->8->128)
// ---------------------------------------------------------------------------
__global__ void se_gate_kernel(const float* __restrict__ xr,
                               const float* __restrict__ w1, const float* __restrict__ b1,
                               const float* __restrict__ gamma, const float* __restrict__ beta,
                               const float* __restrict__ mean, const float* __restrict__ var,
                               const float* __restrict__ w2, const float* __restrict__ b2,
                               float* __restrict__ gate) {
  __shared__ float sw1[8 * 128];
  __shared__ float sw2[128 * 8];
  for (int i = threadIdx.x; i < 1024; i += 256) { sw1[i] = w1[i]; sw2[i] = w2[i]; }
  __syncthreads();
  const long pix = (long)blockIdx.x * 256 + threadIdx.x;  // B*HW
  const int b = (int)(pix >> 14);
  const int n = (int)(pix & 16383);
  const float* xp = xr + (long)b * NC * HW + n;
  float acc[8];
  for (int m = 0; m < 8; ++m) acc[m] = b1[m];
  for (int ci = 0; ci < 128; ++ci) {
    const float xv = xp[(long)ci * HW];
    for (int m = 0; m < 8; ++m) acc[m] += sw1[m * 128 + ci] * xv;
  }
  for (int m = 0; m < 8; ++m) {
    float g = (acc[m] - mean[m]) * __frsqrt_rn(var[m] + 1e-5f) * gamma[m] + beta[m];
    acc[m] = fmaxf(g, 0.0f);
  }
  float* gp = gate + (long)b * NC * HW + n;
  for (int co = 0; co < 128; ++co) {
    float s = b2[co];
    for (int m = 0; m < 8; ++m) s += sw2[co * 8 + m] * acc[m];
    gp[(long)co * HW] = 1.0f / (1.0f + __expf(-s));
  }
}

// ---------------------------------------------------------------------------
extern "C" void kernel_launch(void* const* d_in, const int* in_sizes, int n_in,
                              void* d_out, int out_size, void* d_ws, size_t ws_size,
                              hipStream_t stream) {
  (void)in_sizes; (void)n_in; (void)out_size; (void)ws_size;
  const float* x     = (const float*)d_in[0];
  const float* tempf = (const float*)d_in[1];
  const float* w1    = (const float*)d_in[2];
  const float* b1    = (const float*)d_in[3];
  const float* bng   = (const float*)d_in[4];
  const float* bnb   = (const float*)d_in[5];
  const float* bnm   = (const float*)d_in[6];
  const float* bnv   = (const float*)d_in[7];
  const float* w2    = (const float*)d_in[8];
  const float* b2    = (const float*)d_in[9];
  const float* projf = (const float*)d_in[10];
  const float* temps = (const float*)d_in[11];
  const float* q1w   = (const float*)d_in[12];
  const float* q1b   = (const float*)d_in[13];
  const float* k1w   = (const float*)d_in[14];
  const float* k1b   = (const float*)d_in[15];
  const float* v1w   = (const float*)d_in[16];
  const float* v1b   = (const float*)d_in[17];
  const float* q3    = (const float*)d_in[18];
  const float* q5    = (const float*)d_in[19];
  const float* k3    = (const float*)d_in[20];
  const float* k5    = (const float*)d_in[21];
  const float* v3    = (const float*)d_in[22];
  const float* v5    = (const float*)d_in[23];
  const float* c3    = (const float*)d_in[24];
  const float* c5    = (const float*)d_in[25];
  const float* projs = (const float*)d_in[26];
  float* out = (float*)d_out;

  float* W = (float*)d_ws;
  float* XF_RE = W;             // fft2(x) real
  float* XF_IM = W + NI;        // fft2(x) imag
  float* BA_RE = W + 2 * NI;    // complex scratch / T1 (1x1-conv temp)
  float* BA_IM = W + 3 * NI;
  float* OUTF  = W + 4 * NI;    // |ifft2(attn@qf)|
  float* OUTFL = W + 5 * NI;    // |ifft2(gate*xf)|
  float* QS    = W + 6 * NI;    // spatial q (also GATE earlier)
  float* KS    = W + 7 * NI;
  float* VS    = W + 8 * NI;
  float* OUTS  = W + 9 * NI;
  float* OUTSL = W + 10 * NI;
  float* SMALL = W + 11 * NI;
  float* INVF = SMALL;               // 1024
  float* ATTR = INVF + 1024;         // 64*256
  float* ATTI = ATTR + 16384;        // 64*256
  float* INVQ = ATTI + 16384;        // 1024
  float* INVK = INVQ + 1024;         // 1024
  float* SATT = INVK + 1024;         // 64*256
  float* GATE = QS;                  // overlap: gate consumed before QS written
  float* T1   = BA_RE;               // overlap: BA free after freq branch

  const int LINES = NB * NC * 128;   // 131072

  // ================= frequency branch =================
  // xf = fft2(x)
  fft128_kernel<false, false, false><<<LINES / 4, 256, 0, stream>>>(
      x, nullptr, nullptr, XF_RE, XF_IM, LINES, 128, 1, -1.0f, 1.0f);
  fft128_kernel<true, false, false><<<LINES / 4, 256, 0, stream>>>(
      XF_RE, XF_IM, nullptr, XF_RE, XF_IM, LINES, 1, 128, -1.0f, 1.0f);
  // qn norms, complex self-Gram, complex softmax, attn @ qf
  l2norm_kernel<true><<<1024, 256, 0, stream>>>(XF_RE, XF_IM, INVF);
  gram_wmma<true, true><<<64, 256, 0, stream>>>(XF_RE, XF_IM, nullptr, nullptr,
                                                INVF, INVF, tempf, ATTR, ATTI);
  softmax16_kernel<true><<<4, 256, 0, stream>>>(ATTR, ATTI, 1024);
  attnv_wmma<true><<<dim3(64, 128), 256, 0, stream>>>(ATTR, ATTI, XF_RE, XF_IM,
                                                      BA_RE, BA_IM);
  // |ifft2| over (hd=16, hw=16384)
  ifft16384_kernel<<<1024, 256, 131072, stream>>>(BA_RE, BA_IM);
  ifft16_abs_kernel<<<4096, 256, 0, stream>>>(BA_RE, BA_IM, OUTF);
  // SE gate on xf.real, then |ifft2(gate * xf)|
  se_gate_kernel<<<512, 256, 0, stream>>>(XF_RE, w1, b1, bng, bnb, bnm, bnv, w2, b2,
                                          GATE);
  fft128_kernel<true, true, false><<<LINES / 4, 256, 0, stream>>>(
      XF_RE, XF_IM, GATE, BA_RE, BA_IM, LINES, 128, 1, 1.0f, 1.0f / 128.0f);
  fft128_kernel<true, false, true><<<LINES / 4, 256, 0, stream>>>(
      BA_RE, BA_IM, nullptr, OUTFL, nullptr, LINES, 1, 128, 1.0f, 1.0f / 128.0f);
  // fsa = proj_f(concat(out_f, out_f_l)) -> d_out (store)
  gemm1x1_wmma<true, false, false><<<dim3(NB, 1024), 256, 0, stream>>>(
      projf, nullptr, OUTF, OUTFL, out);

  // ================= spatial branch =================
  gemm1x1_wmma<false, true, false><<<dim3(NB, 1024), 256, 0, stream>>>(
      q1w, q1b, x, nullptr, T1);
  grouped_conv35<<<32768, 256, 0, stream>>>(T1, q3, q5, QS);
  gemm1x1_wmma<false, true, false><<<dim3(NB, 1024), 256, 0, stream>>>(
      k1w, k1b, x, nullptr, T1);
  grouped_conv35<<<32768, 256, 0, stream>>>(T1, k3, k5, KS);
  gemm1x1_wmma<false, true, false><<<dim3(NB, 1024), 256, 0, stream>>>(
      v1w, v1b, x, nullptr, T1);
  grouped_conv35<<<32768, 256, 0, stream>>>(T1, v3, v5, VS);
  grouped_conv35<<<32768, 256, 0, stream>>>(x, c3, c5, OUTSL);

  l2norm_kernel<false><<<1024, 256, 0, stream>>>(QS, nullptr, INVQ);
  l2norm_kernel<false><<<1024, 256, 0, stream>>>(KS, nullptr, INVK);
  gram_wmma<false, false><<<64, 256, 0, stream>>>(QS, nullptr, KS, nullptr,
                                                  INVQ, INVK, temps, SATT, nullptr);
  softmax16_kernel<false><<<4, 256, 0, stream>>>(SATT, nullptr, 1024);
  attnv_wmma<false><<<dim3(64, 128), 256, 0, stream>>>(SATT, nullptr, VS, nullptr,
                                                       OUTS, nullptr);
  // ssa = proj_s(concat(out_s, out_s_l)) -> d_out (accumulate)
  gemm1x1_wmma<true, false, true><<<dim3(NB, 1024), 256, 0, stream>>>(
      projs, nullptr, OUTS, OUTSL, out);
}